// GraphSAGE_5772436045954
// MI455X (gfx1250) — compile-verified
//
#include <hip/hip_runtime.h>

#define N_NODES 100000
#define N_FEAT  128

typedef float v2f __attribute__((ext_vector_type(2)));
typedef float v8f __attribute__((ext_vector_type(8)));

// ---------------------------------------------------------------- utilities
__global__ void zero_f32(float* __restrict__ p, long n) {
  long i = (long)blockIdx.x * blockDim.x + threadIdx.x;
  long stride = (long)gridDim.x * blockDim.x;
  for (; i < n; i += stride) p[i] = 0.0f;
}

__global__ void degree_kernel(const int* __restrict__ dst,
                              float* __restrict__ deg, int ne) {
  int e = blockIdx.x * blockDim.x + threadIdx.x;
  if (e < ne) atomicAdd(deg + dst[e], 1.0f);
}

// One wave per edge: lane g handles features [4g, 4g+4). 512B coalesced gather
// of the source row, 4 f32 atomics into the (L2-resident) agg row.
__global__ void scatter_add(const float* __restrict__ feat,
                            const int* __restrict__ src,
                            const int* __restrict__ dst,
                            float* __restrict__ agg, int ne) {
  long tid = (long)blockIdx.x * blockDim.x + threadIdx.x;
  long e   = tid >> 5;
  int  g   = (int)(tid & 31) * 4;
  if (e >= ne) return;
  int s = src[e], d = dst[e];
  const float4 v = *(const float4*)(feat + (long)s * N_FEAT + g);
  float* b = agg + (long)d * N_FEAT + g;
  atomicAdd(b + 0, v.x);
  atomicAdd(b + 1, v.y);
  atomicAdd(b + 2, v.z);
  atomicAdd(b + 3, v.w);
}

// ------------------------------------------------------- fused SAGE GEMM
// OUT[n, j] = X[n,:]@Wself[:,j] + (AGG[n,:]/max(deg,1))@Wneigh[:,j] + bias[j]
// One wave -> one 16x16 f32 output tile via V_WMMA_F32_16X16X4_F32.
// V_WMMA_F32_16X16X4_F32 operand layout (wave32):
//   A: lane l, vgpr v -> A[l&15][(l>>4)*2 + v]      (16x4)
//   B: lane l, vgpr v -> B[(l>>4)*2 + v][l&15]      (4x16)
//   C/D: lane l, vgpr v -> D[v + (l>>4)*8][l&15]    (16x16)
template <bool RELU>
__global__ __launch_bounds__(256) void sage_gemm(
    const float* __restrict__ X,       // N x 128
    const float* __restrict__ AGG,     // N x 128 (neighbor sums)
    const float* __restrict__ DEG,     // N
    const float* __restrict__ Wself,   // 128 x ncols (row-major)
    const float* __restrict__ Wneigh,  // 128 x ncols
    const float* __restrict__ bias,    // ncols
    float* __restrict__ OUT,           // N x ncols
    int ncols, long ntiles) {
  const int wave = threadIdx.x >> 5;
  const int lane = threadIdx.x & 31;
  const int col_tiles = ncols >> 4;

  long tile = (long)blockIdx.x * 8 + wave;     // wave-uniform
  if (tile >= ntiles) return;                  // whole wave exits: EXEC stays all-1s
  int mt = (int)(tile / col_tiles);
  int ct = (int)(tile % col_tiles);

  const int r  = lane & 15;   // A-row / B-col / D-col within tile
  const int kh = lane >> 4;   // K-half selector

  const long node = (long)mt * 16 + r;
  const int  colg = ct * 16 + r;

  const float* xrow = X   + node * N_FEAT;
  const float* grow = AGG + node * N_FEAT;
  const float invd  = 1.0f / fmaxf(DEG[node], 1.0f);

  v8f c = {0.f, 0.f, 0.f, 0.f, 0.f, 0.f, 0.f, 0.f};

  // self term: X @ Wself
#pragma unroll 8
  for (int k0 = 0; k0 < N_FEAT; k0 += 4) {
    const int k = k0 + kh * 2;                // even -> 8B aligned
    v2f a = *(const v2f*)(xrow + k);
    v2f b;
    b.x = Wself[(long)k * ncols + colg];
    b.y = Wself[(long)(k + 1) * ncols + colg];
    c = __builtin_amdgcn_wmma_f32_16x16x4_f32(false, a, false, b,
                                              (short)0, c, false, false);
  }
  // neighbor term: (AGG * invd_row) @ Wneigh   (mean folded into A load)
#pragma unroll 8
  for (int k0 = 0; k0 < N_FEAT; k0 += 4) {
    const int k = k0 + kh * 2;
    v2f a = *(const v2f*)(grow + k);
    a.x *= invd;
    a.y *= invd;
    v2f b;
    b.x = Wneigh[(long)k * ncols + colg];
    b.y = Wneigh[(long)(k + 1) * ncols + colg];
    c = __builtin_amdgcn_wmma_f32_16x16x4_f32(false, a, false, b,
                                              (short)0, c, false, false);
  }

  const float bb = bias[colg];
  const long rbase = (long)mt * 16 + kh * 8;
#pragma unroll
  for (int v = 0; v < 8; ++v) {
    float val = c[v] + bb;
    if (RELU) val = fmaxf(val, 0.0f);
    OUT[(rbase + v) * ncols + colg] = val;
  }
}

// ---------------------------------------------------------------- driver
extern "C" void kernel_launch(void* const* d_in, const int* in_sizes, int n_in,
                              void* d_out, int out_size, void* d_ws,
                              size_t ws_size, hipStream_t stream) {
  const float* x       = (const float*)d_in[0];
  const int*   src     = (const int*)d_in[1];
  const int*   dst     = (const int*)d_in[2];
  const float* wself0  = (const float*)d_in[3];
  const float* wneigh0 = (const float*)d_in[4];
  const float* b0      = (const float*)d_in[5];
  const float* wself1  = (const float*)d_in[6];
  const float* wneigh1 = (const float*)d_in[7];
  const float* b1      = (const float*)d_in[8];
  const float* wself2  = (const float*)d_in[9];
  const float* wneigh2 = (const float*)d_in[10];
  const float* b2      = (const float*)d_in[11];
  float* out = (float*)d_out;

  const int  ne   = in_sizes[1];
  const long npad = (N_NODES + 63) & ~63L;  // align agg
  float* deg = (float*)d_ws;
  float* agg = deg + npad;
  float* h1  = agg + (long)N_NODES * N_FEAT;
  float* h2  = h1  + (long)N_NODES * N_FEAT;

  const long mtiles = N_NODES / 16;  // 6250 exactly
  const dim3 blk(256);
  const int scat_blocks = (int)(((long)ne * 32 + 255) / 256);
  const int edge_blocks = (ne + 255) / 256;

  // degree (once) + zero agg
  zero_f32<<<2048, blk, 0, stream>>>(deg, npad + (long)N_NODES * N_FEAT);
  degree_kernel<<<edge_blocks, blk, 0, stream>>>(dst, deg, ne);

  // ---- layer 0: 128 -> 128, ReLU
  scatter_add<<<scat_blocks, blk, 0, stream>>>(x, src, dst, agg, ne);
  {
    long ntiles = mtiles * (128 / 16);
    sage_gemm<true><<<(int)(ntiles / 8), blk, 0, stream>>>(
        x, agg, deg, wself0, wneigh0, b0, h1, 128, ntiles);
  }

  // ---- layer 1: 128 -> 128, ReLU
  zero_f32<<<2048, blk, 0, stream>>>(agg, (long)N_NODES * N_FEAT);
  scatter_add<<<scat_blocks, blk, 0, stream>>>(h1, src, dst, agg, ne);
  {
    long ntiles = mtiles * (128 / 16);
    sage_gemm<true><<<(int)(ntiles / 8), blk, 0, stream>>>(
        h1, agg, deg, wself1, wneigh1, b1, h2, 128, ntiles);
  }

  // ---- layer 2: 128 -> 64, no ReLU, write d_out
  zero_f32<<<2048, blk, 0, stream>>>(agg, (long)N_NODES * N_FEAT);
  scatter_add<<<scat_blocks, blk, 0, stream>>>(h2, src, dst, agg, ne);
  {
    long ntiles = mtiles * (64 / 16);
    sage_gemm<false><<<(int)(ntiles / 8), blk, 0, stream>>>(
        h2, agg, deg, wself2, wneigh2, b2, out, 64, ntiles);
  }
}